// GMHLoss_2044404433303
// MI455X (gfx1250) — compile-verified
//
#include <hip/hip_runtime.h>
#include <hip/hip_bf16.h>

#define BINS 10
#define BIN_SCALE_F 9.9999f        /* BINS - 1e-4 */
#define GD_EPS 1e-4f
#define WAVES_PER_BLOCK 8
#define SETS 2                      /* independent LDS accumulator planes per lane */

typedef __attribute__((ext_vector_type(2))) float v2f;
typedef __attribute__((ext_vector_type(8))) float v8f;

// Exact f32 all-lane (wave32) sum using the matrix unit.
// With B = all-ones and C = 0:  D[m][n] = sum_k A[m][k]  (independent of n).
// A 16x4 f32 layout: lanes 0-15 supply K=0,1 (v0,v1); lanes 16-31 supply K=2,3.
// WMMA #1: row m sum = v(lane m) + v(lane m+16); summing the 8 D vgprs in-lane
// gives half-wave totals; WMMA #2 adds the two halves -> full 32-lane sum,
// broadcast to every lane. All f32, no precision loss.
__device__ __forceinline__ float wave_allreduce_f32_wmma(float v) {
  v2f ones; ones[0] = 1.0f; ones[1] = 1.0f;
  v8f c = {};
  v2f a; a[0] = v; a[1] = 0.0f;
  v8f d = __builtin_amdgcn_wmma_f32_16x16x4_f32(
      /*neg_a=*/false, a, /*neg_b=*/false, ones,
      /*c_mod=*/(short)0, c, /*reuse_a=*/false, /*reuse_b=*/false);
  float t = d[0] + d[1] + d[2] + d[3] + d[4] + d[5] + d[6] + d[7];
  v2f a2; a2[0] = t; a2[1] = 0.0f;
  v8f d2 = __builtin_amdgcn_wmma_f32_16x16x4_f32(
      false, a2, false, ones, (short)0, c, false, false);
  return d2[0];
}

// Per-element GHM accumulate into one of this lane's private LDS planes.
// col rows [0..9] = bce sums, [10..19] = counts (gap = 1280B -> fuses into
// ds_{load,store}_2addr_stride64_b32 offset1:5).
__device__ __forceinline__ void ghm_accum(float xx, float tt,
                                          float (*__restrict__ col)[32], int lane) {
  float e  = __expf(-xx);
  float s  = 1.0f / (1.0f + e);              // sigmoid
  float g  = fabsf(s - tt);                  // gradient magnitude, in [0,1]
  int   b  = (int)(g * BIN_SCALE_F);         // floor (g >= 0)
  b = b > (BINS - 1) ? (BINS - 1) : b;
  // stable BCE-with-logits: max(x,0) - x*t + log1p(exp(-|x|))
  float bce = fmaxf(xx, 0.0f) - xx * tt + __logf(1.0f + __expf(-fabsf(xx)));
  col[b][lane]        += bce;
  col[BINS + b][lane] += 1.0f;
}

__global__ void __launch_bounds__(256)
ghm_pass1(const float* __restrict__ x, const float* __restrict__ tg,
          float* __restrict__ ws, int n) {
  // Per-wave, per-lane column histograms: no atomics, no bank conflicts, no
  // races. Two independent planes per lane break the load-add-store LDS
  // dependency chain between consecutive elements.
  __shared__ float hist[WAVES_PER_BLOCK][SETS][2][BINS][32];
  __shared__ float waveTot[WAVES_PER_BLOCK][2 * BINS];

  const int lane = threadIdx.x & 31;
  const int wave = threadIdx.x >> 5;
  float (*__restrict__ col0)[32] = hist[wave][0][0];   // [2*BINS][32]
  float (*__restrict__ col1)[32] = hist[wave][1][0];

#pragma unroll
  for (int b = 0; b < BINS; ++b) {
    col0[b][lane] = 0.0f;        col0[BINS + b][lane] = 0.0f;
    col1[b][lane] = 0.0f;        col1[BINS + b][lane] = 0.0f;
  }

  const int gtid    = blockIdx.x * blockDim.x + threadIdx.x;
  const int gstride = gridDim.x * blockDim.x;
  const int n4      = n >> 2;
  const float4* __restrict__ x4 = (const float4*)x;
  const float4* __restrict__ t4 = (const float4*)tg;

  for (int i = gtid; i < n4; i += gstride) {
    __builtin_prefetch(&x4[i + gstride], 0, 0);   // global_prefetch_b8
    __builtin_prefetch(&t4[i + gstride], 0, 0);
    float4 xv = x4[i];
    float4 tv = t4[i];
    ghm_accum(xv.x, tv.x, col0, lane);   // chain A
    ghm_accum(xv.y, tv.y, col1, lane);   // chain B (independent of A)
    ghm_accum(xv.z, tv.z, col0, lane);
    ghm_accum(xv.w, tv.w, col1, lane);
  }
  // tail (n not multiple of 4)
  for (int i = (n4 << 2) + gtid; i < n; i += gstride)
    ghm_accum(x[i], tg[i], col0, lane);

  // Epilogue: unconditional for all threads -> EXEC all-ones for WMMA.
  // Each lane reads back only its own columns (in-order LDS per wave, no
  // barrier needed), merges the two planes, WMMA-reduces across the wave;
  // result is wave-uniform so lane 0 writes a race-free per-wave slot.
#pragma unroll
  for (int b = 0; b < BINS; ++b) {
    float s = col0[b][lane]        + col1[b][lane];
    float c = col0[BINS + b][lane] + col1[BINS + b][lane];
    float sw = wave_allreduce_f32_wmma(s);
    float cw = wave_allreduce_f32_wmma(c);
    if (lane == 0) {
      waveTot[wave][b]        = sw;
      waveTot[wave][BINS + b] = cw;
    }
  }
  __syncthreads();
  if (threadIdx.x < 2 * BINS) {
    float t = 0.0f;
#pragma unroll
    for (int w = 0; w < WAVES_PER_BLOCK; ++w) t += waveTot[w][threadIdx.x];
    atomicAdd(&ws[threadIdx.x], t);               // global_atomic_add_f32
  }
}

__global__ void ghm_zero(float* __restrict__ ws) {
  if (threadIdx.x < 2 * BINS) ws[threadIdx.x] = 0.0f;
}

// result = sum_b S[b] / max(count[b] * nonempty, eps)   (the 1/N cancels)
__global__ void ghm_finalize(const float* __restrict__ ws, float* __restrict__ out) {
  if (blockIdx.x == 0 && threadIdx.x == 0) {
    float ne = 0.0f;
#pragma unroll
    for (int b = 0; b < BINS; ++b) ne += (ws[BINS + b] > 0.0f) ? 1.0f : 0.0f;
    float r = 0.0f;
#pragma unroll
    for (int b = 0; b < BINS; ++b) {
      float gd = fmaxf(ws[BINS + b] * ne, GD_EPS);
      r += ws[b] / gd;
    }
    out[0] = r;
  }
}

extern "C" void kernel_launch(void* const* d_in, const int* in_sizes, int n_in,
                              void* d_out, int out_size, void* d_ws, size_t ws_size,
                              hipStream_t stream) {
  const float* x  = (const float*)d_in[0];
  const float* tg = (const float*)d_in[1];
  float* ws  = (float*)d_ws;
  float* out = (float*)d_out;
  int n = in_sizes[0];

  ghm_zero<<<1, 32, 0, stream>>>(ws);

  const int threads = 256;                 // 8 waves of 32 on CDNA5
  int n4 = n >> 2;
  // target ~16 float4 (64 elements) per thread; caps global-atomic contention
  long long want = ((long long)n4 + (long long)threads * 16 - 1) /
                   ((long long)threads * 16);
  int blocks = (int)(want < 1 ? 1 : (want > 4096 ? 4096 : want));
  ghm_pass1<<<blocks, threads, 0, stream>>>(x, tg, ws, n);

  ghm_finalize<<<1, 32, 0, stream>>>(ws, out);
}